// SparseMoE_83399674953937
// MI455X (gfx1250) — compile-verified
//
#include <hip/hip_runtime.h>
#include <math.h>

// Problem constants (match reference)
#define DHID  1024
#define NTOK  8192          // B*S = 4*2048
#define NEXP  8
#define TOPKK 4

typedef __attribute__((ext_vector_type(2))) float v2f;
typedef __attribute__((ext_vector_type(8))) float v8f;

// ---------------- workspace layout (units of 4 bytes) ----------------
#define WS_COUNTS 0
#define WS_MASK   8
#define WS_PROB   16
#define WS_TOK    32
#define WS_WT     (32 + NEXP * NTOK)

// ---------------------------------------------------------------------
__global__ void moe_init_ws(int* counts, float* msum, float* psum) {
    int i = threadIdx.x;
    if (i < NEXP) { counts[i] = 0; msum[i] = 0.f; psum[i] = 0.f; }
}

__global__ void moe_zero_out(float* __restrict__ out, long long n) {
    long long i4 = ((long long)blockIdx.x * blockDim.x + threadIdx.x) * 4;
    if (i4 + 3 < n) {
        *(float4*)(out + i4) = make_float4(0.f, 0.f, 0.f, 0.f);
    } else {
        for (int j = 0; j < 4; ++j)
            if (i4 + j < n) out[i4 + j] = 0.f;
    }
}

// ------------------------- router: one wave32 per token ---------------
__global__ __launch_bounds__(128)
void moe_router(const float* __restrict__ x, const float* __restrict__ Wg,
                const float* __restrict__ bg,
                int* counts, float* msum, float* psum,
                int* etok, float* ewt) {
    const int t    = blockIdx.x * 4 + (threadIdx.x >> 5);
    const int lane = threadIdx.x & 31;

    float lg[NEXP];
#pragma unroll
    for (int e = 0; e < NEXP; ++e) lg[e] = 0.f;

    const float* xr = x + (size_t)t * DHID;
    for (int d = lane; d < DHID; d += 32) {
        float xv = xr[d];
        const float* wr = Wg + d * NEXP;
#pragma unroll
        for (int e = 0; e < NEXP; ++e) lg[e] += xv * wr[e];
    }
#pragma unroll
    for (int off = 16; off > 0; off >>= 1) {
#pragma unroll
        for (int e = 0; e < NEXP; ++e) lg[e] += __shfl_xor(lg[e], off, 32);
    }

    if (lane == 0) {
        float mx = -1e30f;
#pragma unroll
        for (int e = 0; e < NEXP; ++e) { lg[e] += bg[e]; mx = fmaxf(mx, lg[e]); }
        float p[NEXP]; float s = 0.f;
#pragma unroll
        for (int e = 0; e < NEXP; ++e) { p[e] = __expf(lg[e] - mx); s += p[e]; }
        float inv = 1.f / s;
#pragma unroll
        for (int e = 0; e < NEXP; ++e) {
            p[e] *= inv;
            atomicAdd(&psum[e], p[e]);
        }
        bool used[NEXP];
#pragma unroll
        for (int e = 0; e < NEXP; ++e) used[e] = false;
        int   isel[TOPKK]; float psel[TOPKK]; float ssel = 0.f;
#pragma unroll
        for (int kk = 0; kk < TOPKK; ++kk) {
            int best = 0; float bv = -1.f;
#pragma unroll
            for (int e = 0; e < NEXP; ++e)
                if (!used[e] && p[e] > bv) { bv = p[e]; best = e; }
            used[best] = true; isel[kk] = best; psel[kk] = bv; ssel += bv;
        }
        float wn = 1.f / (ssel + 1e-6f);
#pragma unroll
        for (int kk = 0; kk < TOPKK; ++kk) {
            int e = isel[kk];
            atomicAdd(&msum[e], 1.f);
            int slot = atomicAdd(&counts[e], 1);
            etok[e * NTOK + slot] = t;
            ewt [e * NTOK + slot] = psel[kk] * wn;
        }
    }
}

// ------------------------ bucketed expert GEMM -------------------------
// Block: 256 threads (8 waves). Tile: BM=64 tokens x BN=256 cols.
// Wave w: rows [32*(w&1), +32), cols [64*(w>>1), +64)
//   -> 2 A-fragments x 4 B-fragments = 8 WMMA accumulators per wave.
// x tile double-buffered in LDS via GLOBAL_LOAD_ASYNC_TO_LDS_B128:
// K-block i+1 is staged asynchronously while computing K-block i.
#define BM 64
#define BN 256
#define KBLK 64
#define XSTRIDE 68    // 64 + 4 pad: LDS bank = (4*M + k) mod 64, conflict-free
#define XBUF (BM * XSTRIDE)

#define ASYNC_CP16(LADDR, GADDR, OFF)                                        \
    asm volatile("global_load_async_to_lds_b128 %0, %1, off offset:" #OFF    \
                 :: "v"(LADDR), "v"(GADDR) : "memory")

__global__ __launch_bounds__(256)
void moe_gemm(const float* __restrict__ x, const float* __restrict__ W,
              const float* __restrict__ bias,
              const int* __restrict__ counts,
              const int* __restrict__ etok, const float* __restrict__ ewt,
              float* __restrict__ out) {
    __shared__ float xs[2 * XBUF];       // 34,816 B (double buffer)
    __shared__ int   tok_s[BM];
    __shared__ float wt_s[BM];

    const int e        = blockIdx.y >> 7;           // 128 row tiles / expert
    const int row_base = (blockIdx.y & 127) * BM;
    const int cnt      = counts[e];
    if (row_base >= cnt) return;                    // uniform early exit

    const int n_tile = blockIdx.x * BN;
    const int tid  = threadIdx.x;
    const int wave = tid >> 5;
    const int lane = tid & 31;
    const int wm2  = wave & 1;                      // M half (32 rows)
    const int wnq  = wave >> 1;                     // N quarter (64 cols)
    const int lhi  = lane >> 4;
    const int llo  = lane & 15;

    if (tid < BM) {
        int r = row_base + tid;
        bool v = (r < cnt);
        tok_s[tid] = v ? etok[e * NTOK + r] : -1;
        wt_s [tid] = v ? ewt [e * NTOK + r] : 0.f;
    }
    __syncthreads();

    // staging assignment: 4 threads per row, 64 B each (4 x b128)
    const int srow  = tid >> 2;
    const int scseg = (tid & 3) * 16;
    const int stok  = tok_s[srow];
    const unsigned s_laddr0 =
        (unsigned)(size_t)(void*)&xs[srow * XSTRIDE + scseg];
    const unsigned s_laddr1 = s_laddr0 + XBUF * 4;
    const float* sgbase = (stok >= 0) ? (x + (size_t)stok * DHID + scseg) : 0;

    v8f acc[2][4];
#pragma unroll
    for (int m = 0; m < 2; ++m)
#pragma unroll
        for (int t = 0; t < 4; ++t)
#pragma unroll
            for (int i = 0; i < 8; ++i) acc[m][t][i] = 0.f;

    const float* Wbase = W + (size_t)e * DHID * DHID;

    // prologue: stage K-block 0 into buffer 0
    if (stok >= 0) {
        const float* g = sgbase;
        ASYNC_CP16(s_laddr0, g, 0);
        ASYNC_CP16(s_laddr0, g, 16);
        ASYNC_CP16(s_laddr0, g, 32);
        ASYNC_CP16(s_laddr0, g, 48);
    }
    asm volatile("s_wait_asynccnt 0x0" ::: "memory");
    __syncthreads();

    for (int ib = 0; ib < DHID / KBLK; ++ib) {
        const int kb = ib * KBLK;

        // stage next K-block into the other buffer (overlaps compute below)
        if (kb + KBLK < DHID && stok >= 0) {
            const float* g = sgbase + kb + KBLK;
            const unsigned l = (ib & 1) ? s_laddr0 : s_laddr1;
            ASYNC_CP16(l, g, 0);
            ASYNC_CP16(l, g, 16);
            ASYNC_CP16(l, g, 32);
            ASYNC_CP16(l, g, 48);
        }

        const float* xs_cur = &xs[(ib & 1) * XBUF];
        const float* wsrc   = Wbase + (size_t)kb * DHID + n_tile + wnq * 64 + llo;
        const float* arow0  = &xs_cur[(wm2 * 32 + llo) * XSTRIDE];
        const float* arow1  = arow0 + 16 * XSTRIDE;
#pragma unroll 4
        for (int k = 0; k < KBLK; k += 4) {
            const int krow = k + 2 * lhi;   // A-frag: lanes16-31 hold K+2,K+3
            v2f a0, a1;
            a0.x = arow0[krow]; a0.y = arow0[krow + 1];
            a1.x = arow1[krow]; a1.y = arow1[krow + 1];
            const float* bp = wsrc + (size_t)krow * DHID;
#pragma unroll
            for (int t = 0; t < 4; ++t) {
                v2f b;
                b.x = bp[t * 16];           // B[krow][n]
                b.y = bp[t * 16 + DHID];    // B[krow+1][n]
                acc[0][t] = __builtin_amdgcn_wmma_f32_16x16x4_f32(
                    false, a0, false, b, (short)0, acc[0][t], false, false);
                acc[1][t] = __builtin_amdgcn_wmma_f32_16x16x4_f32(
                    false, a1, false, b, (short)0, acc[1][t], false, false);
            }
        }

        // next buffer ready + all waves done reading current buffer
        asm volatile("s_wait_asynccnt 0x0" ::: "memory");
        __syncthreads();
    }

    // epilogue: out[token, col] += weight * (acc + bias[e, col])
#pragma unroll
    for (int m = 0; m < 2; ++m) {
#pragma unroll
        for (int t = 0; t < 4; ++t) {
            const int col = n_tile + wnq * 64 + t * 16 + llo;
            const float bv = bias[e * DHID + col];
#pragma unroll
            for (int i = 0; i < 8; ++i) {
                int rl = wm2 * 32 + m * 16 + i + 8 * lhi;  // C/D: M = i + 8*hi
                int tk = tok_s[rl];
                if (tk >= 0) {
                    atomicAdd(&out[(size_t)tk * DHID + col],
                              wt_s[rl] * (acc[m][t][i] + bv));
                }
            }
        }
    }
}

__global__ void moe_aux(const float* msum, const float* psum, float* out) {
    if (threadIdx.x == 0 && blockIdx.x == 0) {
        float a = 0.f;
#pragma unroll
        for (int e = 0; e < NEXP; ++e)
            a += (msum[e] / (float)NTOK) * (psum[e] / (float)NTOK);
        out[(size_t)NTOK * DHID] = a * (float)NEXP;
    }
}

// ---------------------------------------------------------------------
extern "C" void kernel_launch(void* const* d_in, const int* in_sizes, int n_in,
                              void* d_out, int out_size, void* d_ws, size_t ws_size,
                              hipStream_t stream) {
    const float* x  = (const float*)d_in[0];
    const float* Wg = (const float*)d_in[1];
    const float* bg = (const float*)d_in[2];
    const float* W  = (const float*)d_in[3];
    const float* b  = (const float*)d_in[4];
    float* out = (float*)d_out;

    int*   counts = (int*)  d_ws + WS_COUNTS;
    float* msum   = (float*)d_ws + WS_MASK;
    float* psum   = (float*)d_ws + WS_PROB;
    int*   etok   = (int*)  d_ws + WS_TOK;
    float* ewt    = (float*)d_ws + WS_WT;

    moe_init_ws<<<1, 32, 0, stream>>>(counts, msum, psum);

    long long n = (long long)out_size;   // NTOK*DHID + 1
    long long thr = (n + 3) / 4;
    moe_zero_out<<<(unsigned)((thr + 255) / 256), 256, 0, stream>>>(out, n);

    moe_router<<<NTOK / 4, 128, 0, stream>>>(x, Wg, bg, counts, msum, psum, etok, ewt);

    moe_aux<<<1, 32, 0, stream>>>(msum, psum, out);

    dim3 grid(DHID / BN, NEXP * (NTOK / BM));   // 4 x 1024; empty tiles exit early
    moe_gemm<<<grid, 256, 0, stream>>>(x, W, b, counts, etok, ewt, out);
}